// DynamicGCNconv_87093346828457
// MI455X (gfx1250) — compile-verified
//
#include <hip/hip_runtime.h>
#include <math.h>

typedef __attribute__((ext_vector_type(2))) float v2f;
typedef __attribute__((ext_vector_type(8))) float v8f;

// ---------------- degree / normalization ----------------

__global__ void init_deg_kernel(float* deg, int n) {
  int i = blockIdx.x * blockDim.x + threadIdx.x;
  if (i < n) deg[i] = 1.0f;  // self loop contributes 1 to every node's degree
}

__global__ void count_deg_kernel(const long long* __restrict__ ei, float* deg, int E) {
  int e = blockIdx.x * blockDim.x + threadIdx.x;
  if (e < E) atomicAdd(&deg[(int)ei[(size_t)E + e]], 1.0f);  // dst row of edge_index
}

__global__ void rsqrt_kernel(float* deg, int n) {
  int i = blockIdx.x * blockDim.x + threadIdx.x;
  if (i < n) deg[i] = rsqrtf(deg[i]);  // deg >= 1 always (self loop)
}

// ---------------- W pre-pack for WMMA B-fragment b64 loads ----------------
// B fragment needs {W[ka][col], W[ka+1][col]} per lane (stride-Nout pair).
// Pack pairs contiguously so the GEMM hot loop issues one global_load_b64
// per fragment instead of two strided global_load_b32.
__global__ void pack_w_kernel(const float* __restrict__ W, v2f* __restrict__ Wp,
                              int K, int Nout) {
  int i = blockIdx.x * blockDim.x + threadIdx.x;  // over (K/2)*Nout pairs
  int total = (K >> 1) * Nout;
  if (i < total) {
    int kp = i / Nout, col = i - kp * Nout;
    v2f p;
    p.x = W[(size_t)(2 * kp) * Nout + col];
    p.y = W[(size_t)(2 * kp + 1) * Nout + col];
    Wp[i] = p;
  }
}

// ---------------- node-wise GEMM via fp32 WMMA ----------------
// C[M x Nout] = A[M x K] @ W[K x Nout] (W pre-packed as pairs), fp32.
// One wave computes a 16x64 tile: 4 accumulators, A fragment shared.
// M % 16 == 0, K % 4 == 0, Nout % 64 == 0 (holds for 50000, 256, {256,64}).
__global__ void gemm_wmma_f32(const float* __restrict__ A,
                              const v2f* __restrict__ Wp,
                              float* __restrict__ C,
                              int M, int K, int Nout, int ntilesx) {
  int wid  = (blockIdx.x * blockDim.x + threadIdx.x) >> 5;
  int lane = threadIdx.x & 31;
  int total = (M >> 4) * ntilesx;
  if (wid >= total) return;  // wave-uniform exit; WMMA waves keep EXEC all-1s
  int mt = (wid / ntilesx) << 4;
  int nt = (wid % ntilesx) << 6;
  int lhalf = lane >> 4;   // 0: K pair {0,1}, 1: K pair {2,3}
  int lmod  = lane & 15;   // M index for A, N index for B/C

  v8f acc0 = {}, acc1 = {}, acc2 = {}, acc3 = {};
  const float* ap = A + (size_t)(mt + lmod) * K + 2 * lhalf;
  const v2f*   bp = Wp + (size_t)lhalf * Nout + nt + lmod;

  for (int k0 = 0; k0 < K; k0 += 4) {
    // A fragment: lane (lhalf,lmod) holds A[mt+lmod][k0+2*lhalf .. +1]
    v2f a = *(const v2f*)(ap + k0);
    // B fragments: packed pair {W[ka][col], W[ka+1][col]}, ka = k0+2*lhalf
    const v2f* wrow = bp + (size_t)(k0 >> 1) * Nout;
    v2f b0 = wrow[0];
    v2f b1 = wrow[16];
    v2f b2 = wrow[32];
    v2f b3 = wrow[48];
    acc0 = __builtin_amdgcn_wmma_f32_16x16x4_f32(false, a, false, b0, (short)0, acc0, false, false);
    acc1 = __builtin_amdgcn_wmma_f32_16x16x4_f32(false, a, false, b1, (short)0, acc1, false, false);
    acc2 = __builtin_amdgcn_wmma_f32_16x16x4_f32(false, a, false, b2, (short)0, acc2, false, false);
    acc3 = __builtin_amdgcn_wmma_f32_16x16x4_f32(false, a, false, b3, (short)0, acc3, false, false);
  }

  // C/D layout: VGPR v, lanes 0-15 -> M=v, lanes 16-31 -> M=v+8; N = lmod
#pragma unroll
  for (int v = 0; v < 8; ++v) {
    int row = mt + v + 8 * lhalf;
    float* cp = C + (size_t)row * Nout + nt + lmod;
    cp[0]  = acc0[v];
    cp[16] = acc1[v];
    cp[32] = acc2[v];
    cp[48] = acc3[v];
  }
}

// ---------------- aggregation ----------------

// agg[i][:] = h[i][:] * dinv[i]^2   (self-loop message; doubles as zero-init)
__global__ void self_msg_kernel(const float* __restrict__ h, const float* __restrict__ dinv,
                                float* __restrict__ agg, int total, int dshift) {
  int i = blockIdx.x * blockDim.x + threadIdx.x;
  if (i < total) {
    float di = dinv[i >> dshift];
    agg[i] = h[i] * di * di;
  }
}

// one wave per edge; lanes stride the feature dimension (coalesced gather + atomics).
// D templated (256 / 64) so the feature loop fully unrolls.
template <int D>
__global__ void edge_scatter_kernel(const long long* __restrict__ ei,
                                    const float* __restrict__ h,
                                    const float* __restrict__ dinv,
                                    float* agg, int E) {
  int wid  = (blockIdx.x * blockDim.x + threadIdx.x) >> 5;
  int lane = threadIdx.x & 31;
  if (wid >= E) return;
  int s = (int)ei[wid];
  int d = (int)ei[(size_t)E + wid];
  float norm = dinv[s] * dinv[d];
  const float* hs = h + (size_t)s * D;
  float* ad = agg + (size_t)d * D;
#pragma unroll
  for (int j = 0; j < D / 32; ++j)
    atomicAdd(ad + lane + 32 * j, hs[lane + 32 * j] * norm);
}

__global__ void bias_relu_kernel(float* h, const float* __restrict__ b, int total, int dmask) {
  int i = blockIdx.x * blockDim.x + threadIdx.x;
  if (i < total) h[i] = fmaxf(h[i] + b[i & dmask], 0.0f);
}

// ---------------- final bias + log_softmax (64 cols, one wave per row) ----------------

__global__ void bias_logsoftmax_kernel(const float* __restrict__ agg,
                                       const float* __restrict__ b,
                                       float* __restrict__ out, int M) {
  int row  = (blockIdx.x * blockDim.x + threadIdx.x) >> 5;
  int lane = threadIdx.x & 31;
  if (row >= M) return;
  const float* r = agg + (size_t)row * 64;
  float v0 = r[lane] + b[lane];
  float v1 = r[lane + 32] + b[lane + 32];
  float m = fmaxf(v0, v1);
  for (int off = 16; off > 0; off >>= 1)
    m = fmaxf(m, __shfl_xor(m, off, 32));
  float s = expf(v0 - m) + expf(v1 - m);
  for (int off = 16; off > 0; off >>= 1)
    s += __shfl_xor(s, off, 32);
  float ls = logf(s);
  out[(size_t)row * 64 + lane]      = v0 - m - ls;
  out[(size_t)row * 64 + lane + 32] = v1 - m - ls;
}

// ---------------- launch ----------------

extern "C" void kernel_launch(void* const* d_in, const int* in_sizes, int n_in,
                              void* d_out, int out_size, void* d_ws, size_t ws_size,
                              hipStream_t stream) {
  (void)n_in; (void)out_size; (void)ws_size;
  const float*     x  = (const float*)d_in[0];
  const long long* ei = (const long long*)d_in[1];  // int64 edge_index [2, E]
  const float*     W0 = (const float*)d_in[2];
  const float*     b0 = (const float*)d_in[3];
  const float*     W1 = (const float*)d_in[4];
  const float*     b1 = (const float*)d_in[5];
  const float*     W2 = (const float*)d_in[6];
  const float*     b2 = (const float*)d_in[7];
  float* out = (float*)d_out;

  const int N = in_sizes[0] / 256;  // 50000
  const int E = in_sizes[1] / 2;    // 800000

  float* dinv = (float*)d_ws;
  float* bufA = dinv + (size_t)((N + 255) & ~255);
  float* bufB = bufA + (size_t)N * 256;
  v2f*   Wp   = (v2f*)(bufB + (size_t)N * 256);  // 256*256 floats, reused per layer

  const int TB = 256;
  const int mtiles = N >> 4;  // 3125

  // degree + symmetric norm
  init_deg_kernel<<<(N + TB - 1) / TB, TB, 0, stream>>>(dinv, N);
  count_deg_kernel<<<(E + TB - 1) / TB, TB, 0, stream>>>(ei, dinv, E);
  rsqrt_kernel<<<(N + TB - 1) / TB, TB, 0, stream>>>(dinv, N);

  // ---- layer 0: x(256) -> 256, ReLU ----
  {
    int ntx = 256 / 64, waves = mtiles * ntx;
    pack_w_kernel<<<(128 * 256 + TB - 1) / TB, TB, 0, stream>>>(W0, Wp, 256, 256);
    gemm_wmma_f32<<<(waves + 7) / 8, TB, 0, stream>>>(x, Wp, bufA, N, 256, 256, ntx);
    self_msg_kernel<<<(N * 256 + TB - 1) / TB, TB, 0, stream>>>(bufA, dinv, bufB, N * 256, 8);
    edge_scatter_kernel<256><<<(E + 7) / 8, TB, 0, stream>>>(ei, bufA, dinv, bufB, E);
    bias_relu_kernel<<<(N * 256 + TB - 1) / TB, TB, 0, stream>>>(bufB, b0, N * 256, 255);
  }
  // ---- layer 1: 256 -> 256, ReLU ----
  {
    int ntx = 256 / 64, waves = mtiles * ntx;
    pack_w_kernel<<<(128 * 256 + TB - 1) / TB, TB, 0, stream>>>(W1, Wp, 256, 256);
    gemm_wmma_f32<<<(waves + 7) / 8, TB, 0, stream>>>(bufB, Wp, bufA, N, 256, 256, ntx);
    self_msg_kernel<<<(N * 256 + TB - 1) / TB, TB, 0, stream>>>(bufA, dinv, bufB, N * 256, 8);
    edge_scatter_kernel<256><<<(E + 7) / 8, TB, 0, stream>>>(ei, bufA, dinv, bufB, E);
    bias_relu_kernel<<<(N * 256 + TB - 1) / TB, TB, 0, stream>>>(bufB, b1, N * 256, 255);
  }
  // ---- layer 2: 256 -> 64, log_softmax ----
  {
    int ntx = 64 / 64, waves = mtiles * ntx;
    pack_w_kernel<<<(128 * 64 + TB - 1) / TB, TB, 0, stream>>>(W2, Wp, 256, 64);
    gemm_wmma_f32<<<(waves + 7) / 8, TB, 0, stream>>>(bufB, Wp, bufA, N, 256, 64, ntx);
    self_msg_kernel<<<(N * 64 + TB - 1) / TB, TB, 0, stream>>>(bufA, dinv, bufB, N * 64, 6);
    edge_scatter_kernel<64><<<(E + 7) / 8, TB, 0, stream>>>(ei, bufA, dinv, bufB, E);
    bias_logsoftmax_kernel<<<(N + 7) / 8, TB, 0, stream>>>(bufB, b2, out, N);
  }
}